// PoS_GeM_38800734552570
// MI455X (gfx1250) — compile-verified
//
#include <hip/hip_runtime.h>
#include <hip/hip_bf16.h>

typedef __attribute__((ext_vector_type(2))) float v2f;
typedef __attribute__((ext_vector_type(8))) float v8f;

#define BQ 8
#define CQ 256
#define DQ 3
#define QSPLIT 85   // CQ / DQ ; channel blocks [0,85)->d0 [85,170)->d1 [170,256)->d2

__device__ __forceinline__ int chan_dim(int c) {
    int d = c / QSPLIT;
    return d > (DQ - 1) ? (DQ - 1) : d;
}

// Monotone float<->uint encoding so float max/min can use integer atomics.
__device__ __forceinline__ unsigned encf(float f) {
    unsigned u = __float_as_uint(f);
    return (u & 0x80000000u) ? ~u : (u | 0x80000000u);
}
__device__ __forceinline__ float decf(unsigned u) {
    return (u & 0x80000000u) ? __uint_as_float(u & 0x7FFFFFFFu)
                             : __uint_as_float(~u);
}

// ---------------- init: zero acc & wsum, init encoded max/min ----------------
__global__ void k_init(float* __restrict__ acc, long accN,
                       unsigned* __restrict__ smax, unsigned* __restrict__ smin,
                       float* __restrict__ wsum, long bN3, long total) {
    long t = (long)blockIdx.x * blockDim.x + threadIdx.x;
    long stride = (long)gridDim.x * blockDim.x;
    for (; t < total; t += stride) {
        if (t < accN) acc[t] = 0.0f;
        if (t < bN3) {
            smax[t] = 0u;           // encodes "-inf" (below every encoded float)
            smin[t] = 0xFFFFFFFFu;  // encodes "+inf"
            wsum[t] = 0.0f;
        }
    }
}

// ---------------- per-parent coordinate bounds (segment max/min) -------------
__global__ void k_bounds(const float* __restrict__ cc, const float* __restrict__ cn,
                         const int* __restrict__ idx,
                         unsigned* __restrict__ smax, unsigned* __restrict__ smin,
                         int N, int Nn) {
    int t = blockIdx.x * blockDim.x + threadIdx.x;
    if (t >= BQ * N) return;
    int b = t / N;
    int ix = idx[t];
    const float* pc = cc + (long)t * DQ;
    const float* pn = cn + ((long)b * Nn + ix) * DQ;
    long o = ((long)b * Nn + ix) * DQ;
#pragma unroll
    for (int d = 0; d < DQ; ++d) {
        float delta = pc[d] - pn[d];
        unsigned e = encf(delta);
        atomicMax(smax + o + d, e);
        atomicMin(smin + o + d, e);
    }
}

// ---------------- per-point weights + per-parent weight sums -----------------
__global__ void k_weights(const float* __restrict__ cc, const float* __restrict__ cn,
                          const int* __restrict__ idx,
                          const unsigned* __restrict__ smax, const unsigned* __restrict__ smin,
                          float* __restrict__ wpt, float* __restrict__ wsum,
                          int N, int Nn) {
    int t = blockIdx.x * blockDim.x + threadIdx.x;
    if (t >= BQ * N) return;
    int b = t / N;
    int ix = idx[t];
    const float* pc = cc + (long)t * DQ;
    const float* pn = cn + ((long)b * Nn + ix) * DQ;
    long o = ((long)b * Nn + ix) * DQ;
#pragma unroll
    for (int d = 0; d < DQ; ++d) {
        float delta = pc[d] - pn[d];
        float pos = fmaxf(decf(smax[o + d]), 0.0f);
        float neg = fminf(decf(smin[o + d]), 0.0f);
        float inuse = (delta >= 0.0f) ? pos : neg;
        float w = 1.0f - fabsf(delta) / (fabsf(inuse) + 1e-6f);
        wpt[(long)t * DQ + d] = w;
        atomicAdd(wsum + o + d, w);
    }
}

__device__ __forceinline__ float xpow_of(float xv, float p) {
    float xm = fmaxf(xv, 1e-6f);
    return (p == 3.0f) ? xm * xm * xm : powf(xm, p);
}

// ---------------- feature scatter-add, level-0 input layout (B,C,N) ----------
__global__ void k_feat_cn(const float* __restrict__ x, const int* __restrict__ idx,
                          const float* __restrict__ wpt, float* __restrict__ acc,
                          const float* __restrict__ pp, int level,
                          int N, int Nn) {
    long t = (long)blockIdx.x * blockDim.x + threadIdx.x;
    long total = (long)BQ * CQ * N;
    if (t >= total) return;
    int n = (int)(t % N);
    long r = t / N;
    int c = (int)(r % CQ);
    int b = (int)(r / CQ);
    __builtin_prefetch(&x[t + 256], 0, 0);  // global_prefetch_b8
    int ix = idx[(long)b * N + n];
    float w = wpt[((long)b * N + n) * DQ + chan_dim(c)];
    float xp = xpow_of(x[t], pp[level]);
    atomicAdd(acc + ((long)b * Nn + ix) * CQ + c, xp * w);
}

// ---------------- feature scatter-add, interior layout (B,N,C) --------------
// one block (256 threads) per (b,n) row; wave-coalesced atomic bursts
__global__ void k_feat_nc(const float* __restrict__ x, const int* __restrict__ idx,
                          const float* __restrict__ wpt, float* __restrict__ acc,
                          const float* __restrict__ pp, int level,
                          int N, int Nn) {
    int row = blockIdx.x;              // b*N + n
    int c = threadIdx.x;               // 0..255
    int b = row / N;
    int ix = idx[row];
    float w = wpt[(long)row * DQ + chan_dim(c)];
    float xp = xpow_of(x[(long)row * CQ + c], pp[level]);
    atomicAdd(acc + ((long)b * Nn + ix) * CQ + c, xp * w);
}

// ---------------- finalize: (sum_xw / (sum_w+eps))^(1/p), in place ----------
__global__ void k_finalize(float* __restrict__ acc, const float* __restrict__ wsum,
                           const float* __restrict__ pp, int level,
                           int Nn) {
    long t = (long)blockIdx.x * blockDim.x + threadIdx.x;
    long total = (long)BQ * Nn * CQ;
    if (t >= total) return;
    int c = (int)(t % CQ);
    long r = t / CQ;
    int nn = (int)(r % Nn);
    int b = (int)(r / Nn);
    float den = wsum[((long)b * Nn + nn) * DQ + chan_dim(c)] + 1e-6f;
    float v = acc[t] / den;
    float p = pp[level];
    acc[t] = (p == 3.0f) ? cbrtf(v) : powf(v, 1.0f / p);
}

// ---------------- level 3: dense xw tile (B,N,C), N=256 ---------------------
__global__ void k_xw(const float* __restrict__ x, const float* __restrict__ wpt,
                     float* __restrict__ xw, const float* __restrict__ pp,
                     int level, int N) {
    long t = (long)blockIdx.x * blockDim.x + threadIdx.x;
    long total = (long)BQ * N * CQ;
    if (t >= total) return;
    int c = (int)(t % CQ);
    long row = t / CQ;  // b*N + n
    float w = wpt[row * DQ + chan_dim(c)];
    xw[t] = xpow_of(x[t], pp[level]) * w;
}

// ---------------- level 3: WMMA full reduction over 256 points --------------
// D(16x16) = ones(16x4) x B(4x16) chained over K -> every output row is the
// column sum. One wave per (batch, 16-channel tile); 64 chained f32 WMMAs.
__global__ void k_wmma_reduce(const float* __restrict__ xw,
                              const float* __restrict__ wsum,
                              const float* __restrict__ pp,
                              float* __restrict__ out, int N) {
    int tilesPerB = CQ / 16;
    int b = blockIdx.x / tilesPerB;
    int tile = blockIdx.x % tilesPerB;
    int lane = threadIdx.x;            // 0..31, wave32
    int j = lane & 15;                 // column within tile
    int half = lane >> 4;              // 0: rows K=0,1   1: rows K=2,3
    int c0 = tile * 16;

    v8f accv = {};
    v2f av;  av[0] = 1.0f; av[1] = 1.0f;   // A = ones (layout-invariant)
    const float* base = xw + (long)b * N * CQ + c0 + j;

    for (int t = 0; t < N / 4; ++t) {
        int r0 = 4 * t + 2 * half;
        v2f bv;
        bv[0] = base[(long)r0 * CQ];         // K = 2*half + 0
        bv[1] = base[(long)(r0 + 1) * CQ];   // K = 2*half + 1
        accv = __builtin_amdgcn_wmma_f32_16x16x4_f32(
            /*neg_a=*/false, av, /*neg_b=*/false, bv,
            /*c_mod=*/(short)0, accv, /*reuse_a=*/false, /*reuse_b=*/false);
    }

    if (lane < 16) {                   // D row 0 lives in VGPR0, lanes 0..15
        int c = c0 + j;
        float den = wsum[b * DQ + chan_dim(c)] + 1e-6f;
        float v = accv[0] / den;
        float p = pp[3];
        out[(long)b * CQ + c] = (p == 3.0f) ? cbrtf(v) : powf(v, 1.0f / p);
    }
}

extern "C" void kernel_launch(void* const* d_in, const int* in_sizes, int n_in,
                              void* d_out, int out_size, void* d_ws, size_t ws_size,
                              hipStream_t stream) {
    static const int NS[5] = {16384, 4096, 1024, 256, 1};

    const float* x0 = (const float*)d_in[0];
    const int* idxs[4] = {(const int*)d_in[1], (const int*)d_in[2],
                          (const int*)d_in[3], (const int*)d_in[4]};
    const float* coords[5] = {(const float*)d_in[5], (const float*)d_in[6],
                              (const float*)d_in[7], (const float*)d_in[8],
                              (const float*)d_in[9]};
    const float* pp = (const float*)d_in[10];
    float* out = (float*)d_out;

    // workspace layout (floats)
    float* bufA = (float*)d_ws;                                  // 8*4096*256
    float* bufB = bufA + (long)BQ * 4096 * CQ;                   // 8*1024*256
    unsigned* smaxU = (unsigned*)(bufB + (long)BQ * 1024 * CQ);  // 8*4096*3
    unsigned* sminU = smaxU + (long)BQ * 4096 * DQ;
    float* wsum = (float*)(sminU + (long)BQ * 4096 * DQ);
    float* wpt = wsum + (long)BQ * 4096 * DQ;                    // 8*16384*3

    for (int lvl = 0; lvl < 4; ++lvl) {
        int N = NS[lvl], Nn = NS[lvl + 1];
        const float* xin = (lvl == 0) ? x0 : (lvl == 1) ? bufA
                                         : (lvl == 2) ? bufB : bufA;
        float* acc = (lvl == 0) ? bufA : (lvl == 1) ? bufB
                                       : (lvl == 2) ? bufA : nullptr;

        long accN = (lvl < 3) ? (long)BQ * Nn * CQ : 0;
        long bN3 = (long)BQ * Nn * DQ;
        long initTotal = accN > bN3 ? accN : bN3;
        {
            int blocks = (int)((initTotal + 255) / 256);
            k_init<<<blocks, 256, 0, stream>>>(acc, accN, smaxU, sminU, wsum,
                                               bN3, initTotal);
        }
        {
            int threads = BQ * N;
            k_bounds<<<(threads + 255) / 256, 256, 0, stream>>>(
                coords[lvl], coords[lvl + 1], idxs[lvl], smaxU, sminU, N, Nn);
            k_weights<<<(threads + 255) / 256, 256, 0, stream>>>(
                coords[lvl], coords[lvl + 1], idxs[lvl], smaxU, sminU,
                wpt, wsum, N, Nn);
        }
        if (lvl == 0) {
            long total = (long)BQ * CQ * N;
            k_feat_cn<<<(int)((total + 255) / 256), 256, 0, stream>>>(
                xin, idxs[lvl], wpt, acc, pp, lvl, N, Nn);
        } else if (lvl < 3) {
            k_feat_nc<<<BQ * N, CQ, 0, stream>>>(
                xin, idxs[lvl], wpt, acc, pp, lvl, N, Nn);
        } else {
            // dense xw tile into bufB, then WMMA reduction -> d_out
            long total = (long)BQ * N * CQ;
            k_xw<<<(int)((total + 255) / 256), 256, 0, stream>>>(
                xin, wpt, bufB, pp, lvl, N);
            k_wmma_reduce<<<BQ * (CQ / 16), 32, 0, stream>>>(
                bufB, wsum, pp, out, N);
        }
        if (lvl < 3) {
            long total = (long)BQ * Nn * CQ;
            k_finalize<<<(int)((total + 255) / 256), 256, 0, stream>>>(
                acc, wsum, pp, lvl, Nn);
        }
    }
}